// HierarchicalSoftmax_81183471829101
// MI455X (gfx1250) — compile-verified
//
#include <hip/hip_runtime.h>
#include <hip/hip_bf16.h>
#include <math.h>

// Problem constants (match reference)
#define BB      4
#define SS      1024
#define DD      384
#define NITEMS  16384
#define NCLUST  256
#define MCS     128
#define NTOK    (BB * SS)          // 4096
#define NEG_INF (-1.0e9f)

typedef float v2f __attribute__((ext_vector_type(2)));
typedef float v4f __attribute__((ext_vector_type(4)));
typedef float v8f __attribute__((ext_vector_type(8)));

// ---------------------------------------------------------------------------
// Zero the 6 scalar accumulators in workspace.
// ws[0]=sum mask*(-clp)  ws[1]=sum(-clp)  ws[2]=acc hits  ws[3]=sum mask
// ws[4]=sum mask*(-ilp)  ws[5]=sum(-ilp)
// ---------------------------------------------------------------------------
__global__ void hsm_zero_ws(float* __restrict__ ws) {
    if (threadIdx.x < 8) ws[threadIdx.x] = 0.0f;
}

// ---------------------------------------------------------------------------
// Zero-fill dummy_logits (268 MB) with 16B NON-TEMPORAL stores: this region
// exceeds the 192 MB L2, so bypassing allocation (TH=NT) avoids filling L2
// with dirty zero lines. Runs after the compute kernels so the L2 stays
// dedicated to the (25 MB, heavily re-read) embedding tables during compute.
// ---------------------------------------------------------------------------
__global__ __launch_bounds__(256) void hsm_zero_out(float* __restrict__ out) {
    size_t i = (size_t)blockIdx.x * 256u + threadIdx.x;
    v4f z = {};
    __builtin_nontemporal_store(z, (v4f*)out + i);
}

// ---------------------------------------------------------------------------
// Level 1: cluster logits via V_WMMA_F32_16X16X4_F32 (exact fp32), then
// per-token log-softmax over 256 clusters + argmax for accuracy.
// One wave (32 lanes) per 16-token tile; 256 blocks total.
//
// A (16x4 f32): lane L holds M = L%16, K = 2*(L/16) + {0,1}  -> float2 load
// B (4x16 f32): lane L holds N = L%16, K = 2*(L/16) + {0,1}  -> float2 load
// D (16x16 f32): VGPR r, lane L -> M = r + 8*(L/16), N = L%16
// ---------------------------------------------------------------------------
__global__ __launch_bounds__(32) void hsm_cluster_kernel(
    const float* __restrict__ hidden,      // [NTOK, DD]
    const float* __restrict__ cemb,        // [NCLUST, DD]
    const float* __restrict__ item_mask,   // [NTOK]
    const int*   __restrict__ targets,     // [NTOK]
    const int*   __restrict__ cassign,     // [NITEMS]
    float*       __restrict__ ws)
{
    __shared__ float logits[16][NCLUST + 4];   // +4 pad: conflict-free column reads

    const int lane  = threadIdx.x;
    const int row0  = blockIdx.x * 16;         // first token of this tile
    const int m     = lane & 15;
    const int khalf = (lane >> 4) * 2;         // 0 or 2

    const float* aptr = hidden + (size_t)(row0 + m) * DD + khalf;
    const float* bptr = cemb   + (size_t)m          * DD + khalf;

    v8f zero = {};
    v8f acc[16];
#pragma unroll
    for (int n = 0; n < 16; ++n) acc[n] = zero;

    // K loop: 96 steps of 4; B (all 256 clusters, 384 KB) streamed once per
    // wave while 16 accumulator tiles (128 VGPRs) stay resident.
    for (int k = 0; k < DD; k += 4) {
        v2f a = *(const v2f*)(aptr + k);
#pragma unroll
        for (int n = 0; n < 16; ++n) {
            v2f b = *(const v2f*)(bptr + (size_t)n * 16 * DD + k);
            acc[n] = __builtin_amdgcn_wmma_f32_16x16x4_f32(
                /*neg_a=*/false, a, /*neg_b=*/false, b,
                /*c_mod=*/(short)0, acc[n],
                /*reuse_a=*/false, /*reuse_b=*/false);
        }
    }

    // Scatter D tiles to LDS: logits[token_in_tile][cluster]
    const int rowhalf = 8 * (lane >> 4);
#pragma unroll
    for (int n = 0; n < 16; ++n) {
#pragma unroll
        for (int r = 0; r < 8; ++r) {
            logits[r + rowhalf][n * 16 + m] = acc[n][r];
        }
    }
    __syncthreads();

    // log-softmax + argmax: lanes 0..15 each own one token (WMMAs all done,
    // so EXEC divergence here is fine).
    if (lane < 16) {
        const int token = row0 + lane;
        const int tgt   = targets[token];
        const int tc    = cassign[tgt];

        float mx = -3.4e38f;
        int   arg = 0;
        for (int c = 0; c < NCLUST; ++c) {
            float v = logits[lane][c];
            if (v > mx) { mx = v; arg = c; }
        }
        float se = 0.0f;
        for (int c = 0; c < NCLUST; ++c) se += __expf(logits[lane][c] - mx);

        const float clp = logits[lane][tc] - mx - __logf(se);  // target cluster log-prob
        const float msk = item_mask[token];

        atomicAdd(&ws[0], msk * (-clp));
        atomicAdd(&ws[1], -clp);
        atomicAdd(&ws[2], (arg == tc) ? 1.0f : 0.0f);
        atomicAdd(&ws[3], msk);
    }
}

// ---------------------------------------------------------------------------
// Level 2: per-token gather of the target cluster's 128 member embeddings,
// dot with hidden, masked log-softmax, pick target slot.
// One 128-thread block per token (4096 blocks). Embeddings are L2-resident
// (25 MB footprint vs ~0.8 GB of logical reads), so the per-lane row reads
// are served at L2 bandwidth.
// ---------------------------------------------------------------------------
__global__ __launch_bounds__(128) void hsm_member_kernel(
    const float* __restrict__ hidden,      // [NTOK, DD]
    const float* __restrict__ iemb,        // [NITEMS, DD]
    const float* __restrict__ item_mask,   // [NTOK]
    const int*   __restrict__ targets,     // [NTOK]
    const int*   __restrict__ cassign,     // [NITEMS]
    const int*   __restrict__ cindices,    // [NCLUST, MCS]
    const int*   __restrict__ incid,       // [NITEMS]
    float*       __restrict__ ws)
{
    __shared__ __align__(16) float hsh[DD];
    __shared__ float lsh[MCS];
    __shared__ float red[MCS];

    const int t   = blockIdx.x;            // token index
    const int tid = threadIdx.x;           // member slot 0..127
    const int tgt = targets[t];
    const int tc  = cassign[tgt];

    // Stage hidden row in LDS (384 floats).
    for (int i = tid; i < DD; i += MCS) hsh[i] = hidden[(size_t)t * DD + i];
    __syncthreads();

    const int mid = cindices[tc * MCS + tid];
    float logit = NEG_INF;
    if (mid >= 0) {
        const v4f* e = (const v4f*)(iemb + (size_t)mid * DD);
        const v4f* h = (const v4f*)hsh;
        float a0 = 0.0f;
#pragma unroll 4
        for (int i = 0; i < DD / 4; ++i) {
            v4f ev = e[i];
            v4f hv = h[i];
            a0 = fmaf(ev.x, hv.x, a0);
            a0 = fmaf(ev.y, hv.y, a0);
            a0 = fmaf(ev.z, hv.z, a0);
            a0 = fmaf(ev.w, hv.w, a0);
        }
        logit = a0;
    }
    lsh[tid] = logit;
    red[tid] = logit;
    __syncthreads();

    // max reduce over 128
    for (int s = 64; s > 0; s >>= 1) {
        if (tid < s) red[tid] = fmaxf(red[tid], red[tid + s]);
        __syncthreads();
    }
    const float mx = red[0];
    __syncthreads();

    // sum of exp
    red[tid] = __expf(logit - mx);
    __syncthreads();
    for (int s = 64; s > 0; s >>= 1) {
        if (tid < s) red[tid] += red[tid + s];
        __syncthreads();
    }

    if (tid == 0) {
        const float se  = red[0];
        const int   pos = incid[tgt];                       // always a valid slot
        const float ilp = lsh[pos] - mx - __logf(se);       // target item log-prob
        const float msk = item_mask[t];
        atomicAdd(&ws[4], msk * (-ilp));
        atomicAdd(&ws[5], -ilp);
    }
}

// ---------------------------------------------------------------------------
// Combine accumulators into the 4 output scalars.
// ---------------------------------------------------------------------------
__global__ void hsm_finalize(const float* __restrict__ ws,
                             float* __restrict__ out_scalars)
{
    if (threadIdx.x == 0) {
        const float inv_n = 1.0f / (float)NTOK;
        out_scalars[0] = (ws[0] + ws[4]) / (ws[3] + 1e-8f); // total_loss
        out_scalars[1] = ws[1] * inv_n;                     // -mean cluster log-prob
        out_scalars[2] = ws[5] * inv_n;                     // -mean item log-prob
        out_scalars[3] = ws[2] * inv_n;                     // cluster accuracy
    }
}

// ---------------------------------------------------------------------------
extern "C" void kernel_launch(void* const* d_in, const int* in_sizes, int n_in,
                              void* d_out, int out_size, void* d_ws, size_t ws_size,
                              hipStream_t stream) {
    const float* hidden    = (const float*)d_in[0];   // [B,S,D]
    const float* iemb      = (const float*)d_in[1];   // [NITEMS, D]
    const float* cemb      = (const float*)d_in[2];   // [NCLUST, D]
    const float* item_mask = (const float*)d_in[3];   // [B,S]
    const int*   targets   = (const int*)  d_in[4];   // [B,S]
    const int*   cassign   = (const int*)  d_in[5];   // [NITEMS]
    const int*   cindices  = (const int*)  d_in[6];   // [NCLUST, MCS]
    const int*   incid     = (const int*)  d_in[7];   // [NITEMS]

    float* out = (float*)d_out;
    float* ws  = (float*)d_ws;

    const size_t dummy_elems = (size_t)NTOK * (size_t)NITEMS;  // 67108864

    // Compute first (clean L2 for embedding reuse), stream the 268 MB of
    // zeros last with NT stores (pure HBM-bandwidth phase, no L2 pollution).
    hsm_zero_ws<<<1, 32, 0, stream>>>(ws);
    hsm_cluster_kernel<<<NTOK / 16, 32, 0, stream>>>(hidden, cemb, item_mask,
                                                     targets, cassign, ws);
    hsm_member_kernel<<<NTOK, 128, 0, stream>>>(hidden, iemb, item_mask, targets,
                                                cassign, cindices, incid, ws);
    hsm_finalize<<<1, 32, 0, stream>>>(ws, out + dummy_elems);
    hsm_zero_out<<<(unsigned)(dummy_elems / (4 * 256)), 256, 0, stream>>>(out);
}